// GNN_83047487635729
// MI455X (gfx1250) — compile-verified
//
#include <hip/hip_runtime.h>
#include <hip/hip_bf16.h>

#define N_NODES  50000
#define N_EDGES  800000
#define DIM      128
#define N_LAYERS 3
#define N_GRAPHS 64
#define E_TOT    (N_EDGES + N_NODES)   // self-loops appended
#define NEG_SLOPE 0.2f

#define MT       5                     // M-tiles (16 rows each) per block
#define ROWS     (MT * 16)             // 80 rows/block; 50000/80 = 625 exactly
#define LDSP     132                   // padded row stride (kills bank conflicts)

typedef float v2f __attribute__((ext_vector_type(2)));
typedef float v8f __attribute__((ext_vector_type(8)));

// ---- ordered-uint encoding for float atomicMax (handles negatives) ----
__device__ __forceinline__ unsigned fenc(float f) {
    unsigned u = __float_as_uint(f);
    return (u & 0x80000000u) ? ~u : (u | 0x80000000u);
}
__device__ __forceinline__ float fdec(unsigned u) {
    return __uint_as_float((u & 0x80000000u) ? (u ^ 0x80000000u) : ~u);
}
#define ENC_NEG_INF 0x007FFFFFu   // fenc(-inf)

// =====================================================================
// H[50000x128] = X[50000x128] @ W[128x128], fp32 WMMA 16x16x4.
// Block = 256 thr = 8 waves, computes an 80x128 stripe of H.
//  - X stripe (80x128) and W^T (128x128) staged in LDS, rows padded to 132.
//  - k-loop outer: one B pair (ds_load_b64) feeds 5 WMMAs (5 M-tiles),
//    so the stream is WMMA-dense instead of vmem-dense.
// =====================================================================
__global__ __launch_bounds__(256) void k_gemm(const float* __restrict__ X,
                                              const float* __restrict__ W,
                                              float* __restrict__ H) {
    __shared__ float ldsX[ROWS * LDSP];    // 80*132*4  = 42240 B
    __shared__ float ldsWt[DIM * LDSP];    // 128*132*4 = 67584 B  (Wt[n][k]=W[k][n])
    const int tid = threadIdx.x;
    const int m0  = blockIdx.x * ROWS;

    // stage X stripe: 80x128 = 2560 float4, coalesced
    {
        const float4* gx = reinterpret_cast<const float4*>(X + (size_t)m0 * DIM);
        for (int i = tid; i < ROWS * 32; i += 256) {
            const int row = i >> 5, c4 = (i & 31) * 4;
            *reinterpret_cast<float4*>(&ldsX[row * LDSP + c4]) = gx[i];
        }
    }
    // stage W transposed: coalesced float4 global reads, scattered LDS writes
    {
        const float4* gw = reinterpret_cast<const float4*>(W);
        for (int i = tid; i < DIM * 32; i += 256) {
            const int k = i >> 5, n = (i & 31) * 4;
            const float4 v = gw[i];
            ldsWt[(n + 0) * LDSP + k] = v.x;
            ldsWt[(n + 1) * LDSP + k] = v.y;
            ldsWt[(n + 2) * LDSP + k] = v.z;
            ldsWt[(n + 3) * LDSP + k] = v.w;
        }
    }
    __syncthreads();

    const int wave = tid >> 5;
    const int lane = tid & 31;
    const int n0   = wave * 16;
    const int nidx = lane & 15;            // N (for B/D) or M (for A) within tile
    const int koff = (lane >> 4) << 1;     // lanes 16-31 carry K+2, K+3

    v8f acc[MT] = {};
    const float* bw = &ldsWt[(n0 + nidx) * LDSP];

    for (int k = 0; k < DIM; k += 4) {
        v2f b;
        b.x = bw[k + koff];
        b.y = bw[k + koff + 1];
        #pragma unroll
        for (int t = 0; t < MT; ++t) {
            const float* ax = &ldsX[(t * 16 + nidx) * LDSP];
            v2f a;
            a.x = ax[k + koff];
            a.y = ax[k + koff + 1];
            acc[t] = __builtin_amdgcn_wmma_f32_16x16x4_f32(false, a, false, b,
                                                           (short)0, acc[t],
                                                           false, false);
        }
    }

    // D layout: VGPR r, lanes 0-15 -> (M=r, N=lane); lanes 16-31 -> (M=r+8, N=lane-16)
    const int col  = n0 + nidx;
    const int rsub = (lane >> 4) << 3;
    #pragma unroll
    for (int t = 0; t < MT; ++t) {
        const int rbase = m0 + t * 16 + rsub;
        #pragma unroll
        for (int r = 0; r < 8; ++r)
            H[(size_t)(rbase + r) * DIM + col] = acc[t][r];
    }
}

// =====================================================================
// Per-node attention scores: s_src[n] = h[n]·a_src ; s_dst[n] = h[n]·a_dst
// One wave per node, float4 lanes, shfl-xor reduction (wave32).
// =====================================================================
__global__ __launch_bounds__(256) void k_dots(const float* __restrict__ H,
                                              const float* __restrict__ a_src,
                                              const float* __restrict__ a_dst,
                                              float* __restrict__ s_src,
                                              float* __restrict__ s_dst) {
    const int gw   = (blockIdx.x * blockDim.x + threadIdx.x) >> 5;
    const int lane = threadIdx.x & 31;
    if (gw >= N_NODES) return;
    const float4 hv = reinterpret_cast<const float4*>(H + (size_t)gw * DIM)[lane];
    const float4 as = reinterpret_cast<const float4*>(a_src)[lane];
    const float4 ad = reinterpret_cast<const float4*>(a_dst)[lane];
    float ps = hv.x * as.x + hv.y * as.y + hv.z * as.z + hv.w * as.w;
    float pd = hv.x * ad.x + hv.y * ad.y + hv.z * ad.z + hv.w * ad.w;
    #pragma unroll
    for (int off = 16; off > 0; off >>= 1) {
        ps += __shfl_xor(ps, off, 32);
        pd += __shfl_xor(pd, off, 32);
    }
    if (lane == 0) { s_src[gw] = ps; s_dst[gw] = pd; }
}

__global__ __launch_bounds__(256) void k_init_nodes(unsigned* __restrict__ m_enc,
                                                    float* __restrict__ ssum) {
    const int i = blockIdx.x * blockDim.x + threadIdx.x;
    if (i < N_NODES) { m_enc[i] = ENC_NEG_INF; ssum[i] = 0.f; }
}

__global__ __launch_bounds__(256) void k_zero(float* __restrict__ p, int n) {
    const int i = blockIdx.x * blockDim.x + threadIdx.x;
    if (i < n) p[i] = 0.f;
}

__device__ __forceinline__ void edge_sd(const int* __restrict__ ei, int e,
                                        int& s, int& d) {
    if (e < N_EDGES) { s = ei[e]; d = ei[N_EDGES + e]; }
    else             { s = d = e - N_EDGES; }          // self-loop
}

// pass 1: alpha = leaky_relu(s_src[src]+s_dst[dst]); segment-max into m_enc[dst]
__global__ __launch_bounds__(256) void k_edge_alpha(const int* __restrict__ ei,
                                                    const float* __restrict__ s_src,
                                                    const float* __restrict__ s_dst,
                                                    float* __restrict__ alpha,
                                                    unsigned* __restrict__ m_enc) {
    const int e = blockIdx.x * blockDim.x + threadIdx.x;
    if (e >= E_TOT) return;
    int s, d; edge_sd(ei, e, s, d);
    float a = s_src[s] + s_dst[d];
    a = (a > 0.f) ? a : NEG_SLOPE * a;
    alpha[e] = a;
    atomicMax(&m_enc[d], fenc(a));
}

// pass 2: e = exp(alpha - m[dst]); segment-sum into ssum[dst]; alpha := e
__global__ __launch_bounds__(256) void k_edge_exp(const int* __restrict__ ei,
                                                  const unsigned* __restrict__ m_enc,
                                                  float* __restrict__ alpha,
                                                  float* __restrict__ ssum) {
    const int e = blockIdx.x * blockDim.x + threadIdx.x;
    if (e >= E_TOT) return;
    int s, d; edge_sd(ei, e, s, d);
    const float ex = __expf(alpha[e] - fdec(m_enc[d]));
    alpha[e] = ex;
    atomicAdd(&ssum[d], ex);
}

// pass 3: out[dst] += (e / sum[dst]) * h[src]   (wave per edge, float4 lanes)
__global__ __launch_bounds__(256) void k_edge_scatter(const int* __restrict__ ei,
                                                      const float* __restrict__ alpha,
                                                      const float* __restrict__ ssum,
                                                      const float* __restrict__ H,
                                                      float* __restrict__ out) {
    const int gw   = (blockIdx.x * blockDim.x + threadIdx.x) >> 5;
    const int lane = threadIdx.x & 31;
    if (gw >= E_TOT) return;
    int s, d; edge_sd(ei, gw, s, d);
    const float w = alpha[gw] / (ssum[d] + 1e-16f);
    const float4 hv = reinterpret_cast<const float4*>(H + (size_t)s * DIM)[lane];
    float* o = out + (size_t)d * DIM + lane * 4;
    atomicAdd(o + 0, w * hv.x);
    atomicAdd(o + 1, w * hv.y);
    atomicAdd(o + 2, w * hv.z);
    atomicAdd(o + 3, w * hv.w);
}

__global__ __launch_bounds__(256) void k_bias_relu(float* __restrict__ x,
                                                   const float* __restrict__ b) {
    const int i = blockIdx.x * blockDim.x + threadIdx.x;
    if (i >= N_NODES * DIM) return;
    const float v = x[i] + b[i & (DIM - 1)];
    x[i] = v > 0.f ? v : 0.f;
}

__global__ __launch_bounds__(256) void k_pool(const float* __restrict__ x,
                                              const int* __restrict__ batch,
                                              float* __restrict__ pooled) {
    const int i = blockIdx.x * blockDim.x + threadIdx.x;
    if (i >= N_NODES * DIM) return;
    const int n = i >> 7, d = i & (DIM - 1);
    atomicAdd(&pooled[batch[n] * DIM + d], x[i]);
}

// y[g] = pooled[g]·Wf + bf  (wave per graph)
__global__ __launch_bounds__(256) void k_final(const float* __restrict__ pooled,
                                               const float* __restrict__ Wf,
                                               const float* __restrict__ bf,
                                               float* __restrict__ y) {
    const int gw   = (blockIdx.x * blockDim.x + threadIdx.x) >> 5;
    const int lane = threadIdx.x & 31;
    if (gw >= N_GRAPHS) return;
    const float4 p = reinterpret_cast<const float4*>(pooled + (size_t)gw * DIM)[lane];
    const float4 w = reinterpret_cast<const float4*>(Wf)[lane];
    float s = p.x * w.x + p.y * w.y + p.z * w.z + p.w * w.w;
    #pragma unroll
    for (int off = 16; off > 0; off >>= 1) s += __shfl_xor(s, off, 32);
    if (lane == 0) y[gw] = s + bf[0];
}

// =====================================================================
extern "C" void kernel_launch(void* const* d_in, const int* in_sizes, int n_in,
                              void* d_out, int out_size, void* d_ws, size_t ws_size,
                              hipStream_t stream) {
    (void)in_sizes; (void)n_in; (void)out_size; (void)ws_size;

    const float* x     = (const float*)d_in[0];
    const int*   ei    = (const int*)d_in[1];
    const int*   batch = (const int*)d_in[2];
    const float* Ws    = (const float*)d_in[3];
    const float* asrc  = (const float*)d_in[4];
    const float* adst  = (const float*)d_in[5];
    const float* bias  = (const float*)d_in[6];
    const float* Wf    = (const float*)d_in[7];
    const float* bf    = (const float*)d_in[8];
    float*       out   = (float*)d_out;

    // ---- workspace carve-up (all 16B-aligned slices) ----
    float* ws = (float*)d_ws;
    float*    bufA   = ws;            ws += (size_t)N_NODES * DIM;  // h = x@W
    float*    bufB   = ws;            ws += (size_t)N_NODES * DIM;  // aggregated out
    float*    alpha  = ws;            ws += E_TOT;
    float*    s_src  = ws;            ws += N_NODES;
    float*    s_dst  = ws;            ws += N_NODES;
    float*    ssum   = ws;            ws += N_NODES;
    unsigned* m_enc  = (unsigned*)ws; ws += N_NODES;
    float*    pooled = ws;            ws += N_GRAPHS * DIM;

    const int NB256     = 256;
    const int g_nodes   = (N_NODES + 255) / 256;
    const int g_feat    = (N_NODES * DIM + 255) / 256;
    const int g_edges   = (E_TOT + 255) / 256;
    const int g_edgewav = (int)(((size_t)E_TOT * 32 + 255) / 256);
    const int g_nodewav = (N_NODES * 32 + 255) / 256;

    const float* xcur = x;
    for (int l = 0; l < N_LAYERS; ++l) {
        const float* W  = Ws   + (size_t)l * DIM * DIM;
        const float* as = asrc + (size_t)l * DIM;
        const float* ad = adst + (size_t)l * DIM;
        const float* b  = bias + (size_t)l * DIM;

        k_gemm<<<N_NODES / ROWS, NB256, 0, stream>>>(xcur, W, bufA);
        k_dots<<<g_nodewav, NB256, 0, stream>>>(bufA, as, ad, s_src, s_dst);
        k_init_nodes<<<g_nodes, NB256, 0, stream>>>(m_enc, ssum);
        k_zero<<<g_feat, NB256, 0, stream>>>(bufB, N_NODES * DIM);
        k_edge_alpha<<<g_edges, NB256, 0, stream>>>(ei, s_src, s_dst, alpha, m_enc);
        k_edge_exp<<<g_edges, NB256, 0, stream>>>(ei, m_enc, alpha, ssum);
        k_edge_scatter<<<g_edgewav, NB256, 0, stream>>>(ei, alpha, ssum, bufA, bufB);
        k_bias_relu<<<g_feat, NB256, 0, stream>>>(bufB, b);
        xcur = bufB;
    }

    k_zero<<<(N_GRAPHS * DIM + 255) / 256, NB256, 0, stream>>>(pooled, N_GRAPHS * DIM);
    k_pool<<<g_feat, NB256, 0, stream>>>(bufB, batch, pooled);
    k_final<<<(N_GRAPHS * 32 + 255) / 256, NB256, 0, stream>>>(pooled, Wf, bf, out);
}